// CombinedLoss_39152921870911
// MI455X (gfx1250) — compile-verified
//
#include <hip/hip_runtime.h>
#include <hip/hip_bf16.h>
#include <math.h>

// ---------------------------------------------------------------------------
// CombinedLoss (Dice 0.7 + Sobel-boundary BCE 0.3) for [8,4,512,512] f32.
// Single streaming pass: sigmoid + 3x3 Sobel stencil + reductions.
// CDNA5 paths: async global->LDS tile loads (ASYNCcnt), WMMA f32 16x16x4
// wave reductions. Deterministic (no float atomics).
// ---------------------------------------------------------------------------

#define B_ 8
#define C_ 4
#define H_ 512
#define W_ 512
#define PLANES (B_ * C_)          // 32
#define PLANE_ELEMS (H_ * W_)     // 262144
#define TILE 64
#define HALO 66                   // TILE + 2
#define NHALO (HALO * HALO)       // 4356
#define TPB 256
#define TILES_X (W_ / TILE)       // 8
#define TILES_Y (H_ / TILE)       // 8
#define BLOCKS_PER_PLANE (TILES_X * TILES_Y)   // 64
#define NBLOCKS (BLOCKS_PER_PLANE * PLANES)    // 2048
#define TOTAL_ELEMS 8388608.0f    // B*C*H*W

typedef float v2f __attribute__((ext_vector_type(2)));
typedef float v8f __attribute__((ext_vector_type(8)));

#if defined(__has_builtin)
#  if __has_builtin(__builtin_amdgcn_wmma_f32_16x16x4_f32)
#    define USE_WMMA_REDUCE 1
#  endif
#  if __has_builtin(__builtin_amdgcn_global_load_async_to_lds_b32)
#    define USE_ASYNC_LDS 1
#  endif
#endif
#ifndef USE_WMMA_REDUCE
#  define USE_WMMA_REDUCE 0
#endif
#ifndef USE_ASYNC_LDS
#  define USE_ASYNC_LDS 0
#endif

#define AS_GLOBAL_I32(p) ((__attribute__((address_space(1))) int*)(void*)(p))
#define AS_LDS_I32(p)    ((__attribute__((address_space(3))) int*)(void*)(p))

// Full-wave (32 lane) f32 sum.
// WMMA trick: A(16x4) with a.x = v, a.y = 0 -> per ISA layout A[m][0]=v[m]
// (lanes 0-15) and A[m][2]=v[m+16] (lanes 16-31). B = all ones (layout-proof).
// D[m][n] = v[m] + v[m+16]. Lane n holds D rows {0..7} (n<16) or {8..15}
// (n>=16) in 8 VGPRs; summing them + xor-16 shuffle = full wave sum, exact f32.
__device__ __forceinline__ float wave_sum(float v) {
#if USE_WMMA_REDUCE
    v2f a; a.x = v;    a.y = 0.0f;
    v2f b; b.x = 1.0f; b.y = 1.0f;
    v8f c = {};
    v8f d = __builtin_amdgcn_wmma_f32_16x16x4_f32(
        false, a, false, b, (short)0, c, false, false);
    float s = ((d[0] + d[1]) + (d[2] + d[3])) + ((d[4] + d[5]) + (d[6] + d[7]));
    s += __shfl_xor(s, 16, 32);
    return s;
#else
    for (int m = 16; m > 0; m >>= 1) v += __shfl_xor(v, m, 32);
    return v;
#endif
}

__global__ __launch_bounds__(TPB)
void combined_loss_tiles(const float* __restrict__ logits,
                         const float* __restrict__ targets,
                         float* __restrict__ ws) {
    __shared__ float Pt[NHALO];   // sigmoid(logits) tile + halo (0 outside image)
    __shared__ float Tt[NHALO];   // targets tile + halo (0 outside image)
    __shared__ float red[8][4];

    const int tid = threadIdx.x;
    const int x0 = blockIdx.x * TILE;
    const int y0 = blockIdx.y * TILE;
    const int pz = blockIdx.z;                       // plane = b*C + c
    const float* __restrict__ gl = logits  + (size_t)pz * PLANE_ELEMS;
    const float* __restrict__ gt = targets + (size_t)pz * PLANE_ELEMS;

    // ---- Stage 1: stream raw tiles into LDS (async path bypasses VGPRs) ----
    for (int e = tid; e < NHALO; e += TPB) {
        const int hy = e / HALO;
        const int hx = e - hy * HALO;
        const int gy = y0 + hy - 1;
        const int gx = x0 + hx - 1;
        const bool inb = ((unsigned)gy < (unsigned)H_) & ((unsigned)gx < (unsigned)W_);
        if (inb) {
            const int off = (gy << 9) + gx;
#if USE_ASYNC_LDS
            __builtin_amdgcn_global_load_async_to_lds_b32(
                AS_GLOBAL_I32(gl + off), AS_LDS_I32(&Pt[e]), 0, 0);
            __builtin_amdgcn_global_load_async_to_lds_b32(
                AS_GLOBAL_I32(gt + off), AS_LDS_I32(&Tt[e]), 0, 0);
#else
            Pt[e] = gl[off];
            Tt[e] = gt[off];
#endif
        }
    }
#if USE_ASYNC_LDS
#  if __has_builtin(__builtin_amdgcn_s_wait_asynccnt)
    __builtin_amdgcn_s_wait_asynccnt(0);
#  else
    asm volatile("s_wait_asynccnt 0" ::: "memory");
#  endif
#endif
    __syncthreads();

    // ---- Stage 2: in-LDS sigmoid; enforce zero padding (SAME conv) ----------
    for (int e = tid; e < NHALO; e += TPB) {
        const int hy = e / HALO;
        const int hx = e - hy * HALO;
        const int gy = y0 + hy - 1;
        const int gx = x0 + hx - 1;
        const bool inb = ((unsigned)gy < (unsigned)H_) & ((unsigned)gx < (unsigned)W_);
        const float lv = Pt[e];
        Pt[e] = inb ? (1.0f / (1.0f + __expf(-lv))) : 0.0f;   // pad=0, NOT sigmoid(0)
        if (!inb) Tt[e] = 0.0f;
    }
    __syncthreads();

    // ---- Stage 3: sliding-window Sobel + accumulators -----------------------
    // Thread = column lx, 16 consecutive rows; 3 new LDS reads/array/pixel.
    const int lx  = tid & 63;
    const int ly0 = (tid >> 6) << 4;
    int base = ly0 * HALO + lx;

    float p00 = Pt[base],        p01 = Pt[base + 1],        p02 = Pt[base + 2];
    float p10 = Pt[base + HALO], p11 = Pt[base + HALO + 1], p12 = Pt[base + HALO + 2];
    float t00 = Tt[base],        t01 = Tt[base + 1],        t02 = Tt[base + 2];
    float t10 = Tt[base + HALO], t11 = Tt[base + HALO + 1], t12 = Tt[base + HALO + 2];

    float inter = 0.0f, psum = 0.0f, tsum = 0.0f, mism = 0.0f;
    int rowOff = base + 2 * HALO;
#pragma unroll
    for (int r = 0; r < 16; ++r) {
        const float p20 = Pt[rowOff], p21 = Pt[rowOff + 1], p22 = Pt[rowOff + 2];
        const float t20 = Tt[rowOff], t21 = Tt[rowOff + 1], t22 = Tt[rowOff + 2];

        const float gxp = (p02 - p00) + 2.0f * (p12 - p10) + (p22 - p20);
        const float gyp = (p20 - p00) + 2.0f * (p21 - p01) + (p22 - p02);
        const bool  pb  = (gxp * gxp + gyp * gyp) > 0.25f;   // |g| > 0.5

        const float gxt = (t02 - t00) + 2.0f * (t12 - t10) + (t22 - t20);
        const float gyt = (t20 - t00) + 2.0f * (t21 - t01) + (t22 - t02);
        const bool  tb  = (gxt * gxt + gyt * gyt) > 0.25f;

        mism  += (pb != tb) ? 1.0f : 0.0f;     // BCE on binary maps = 100*mismatch
        inter += p11 * t11;
        psum  += p11;
        tsum  += t11;

        p00 = p10; p01 = p11; p02 = p12; p10 = p20; p11 = p21; p12 = p22;
        t00 = t10; t01 = t11; t02 = t12; t10 = t20; t11 = t21; t12 = t22;
        rowOff += HALO;
    }

    // ---- Stage 4: deterministic block reduction (WMMA wave sums) ------------
    const float r0 = wave_sum(inter);
    const float r1 = wave_sum(psum);
    const float r2 = wave_sum(tsum);
    const float r3 = wave_sum(mism);

    const int wave = tid >> 5;
    if ((tid & 31) == 0) {
        red[wave][0] = r0; red[wave][1] = r1; red[wave][2] = r2; red[wave][3] = r3;
    }
    __syncthreads();
    if (tid == 0) {
        float a = 0.0f, b = 0.0f, c = 0.0f, m = 0.0f;
#pragma unroll
        for (int w = 0; w < 8; ++w) {
            a += red[w][0]; b += red[w][1]; c += red[w][2]; m += red[w][3];
        }
        const int bid = (blockIdx.z << 6) + (blockIdx.y << 3) + blockIdx.x;
        float* o = ws + (size_t)bid * 4;
        o[0] = a; o[1] = b; o[2] = c; o[3] = m;
    }
}

__global__ __launch_bounds__(32)
void combined_loss_finalize(const float* __restrict__ ws,
                            const float* __restrict__ cw,
                            float* __restrict__ out) {
    __shared__ float sd[PLANES];
    __shared__ float sm[PLANES];
    const int pz = threadIdx.x;            // one plane per lane
    float inter = 0.0f, ps = 0.0f, ts = 0.0f, mm = 0.0f;
    for (int i = 0; i < BLOCKS_PER_PLANE; ++i) {
        const float* p = ws + (size_t)(pz * BLOCKS_PER_PLANE + i) * 4;
        inter += p[0]; ps += p[1]; ts += p[2]; mm += p[3];
    }
    sd[pz] = (2.0f * inter + 1.0f) / (ps + ts + 1.0f);   // dice[b,c], SMOOTH=1
    sm[pz] = mm;
    __syncthreads();
    if (pz == 0) {
        float wsum = 0.0f, dl = 0.0f, mtot = 0.0f;
        for (int c = 0; c < C_; ++c) {
            float md = 0.0f;
            for (int b = 0; b < B_; ++b) md += sd[b * C_ + c];
            md *= (1.0f / (float)B_);
            dl += cw[c] * (1.0f - md);
            wsum += cw[c];
        }
        for (int i = 0; i < PLANES; ++i) mtot += sm[i];
        const float dice_loss     = dl / wsum;
        const float boundary_loss = 100.0f * mtot / TOTAL_ELEMS;
        out[0] = 0.7f * dice_loss + 0.3f * boundary_loss;
    }
}

extern "C" void kernel_launch(void* const* d_in, const int* in_sizes, int n_in,
                              void* d_out, int out_size, void* d_ws, size_t ws_size,
                              hipStream_t stream) {
    (void)in_sizes; (void)n_in; (void)out_size; (void)ws_size;
    const float* logits  = (const float*)d_in[0];
    const float* targets = (const float*)d_in[1];
    const float* cw      = (const float*)d_in[2];
    float* out = (float*)d_out;
    float* ws  = (float*)d_ws;   // 2048 blocks * 4 floats = 32 KB, fully overwritten

    dim3 grid(TILES_X, TILES_Y, PLANES);
    combined_loss_tiles<<<grid, TPB, 0, stream>>>(logits, targets, ws);
    combined_loss_finalize<<<1, 32, 0, stream>>>(ws, cw, out);
}